// ScaleDotProductAttention_58789512348388
// MI455X (gfx1250) — compile-verified
//
#include <hip/hip_runtime.h>

typedef float v2f __attribute__((ext_vector_type(2)));
typedef float v8f __attribute__((ext_vector_type(8)));

#define SEQ 8192
#define DIM 64

// One block = 32 query rows. 8 waves = 2 row-tiles (mt) x 4 column-slices (ct).
// Phase A: accumulate rowsum(exp(S)) and O = exp(S) @ V over all 8192 keys (WMMA f32).
// Phase B: recompute S, write normalized softmax probabilities directly (single pass
// over the 256MB score matrix -> bandwidth floor ~11us at 23.3 TB/s).
__global__ __launch_bounds__(256)
void attn_fused_kernel(const float* __restrict__ Q,
                       const float* __restrict__ K,
                       const float* __restrict__ V,
                       float* __restrict__ Out,
                       float* __restrict__ Score)
{
    __shared__ float eT[8][16 * 18];        // per-wave E-tile staging (pitch 18: conflict-free)
    __shared__ float rsWG[4][32];           // rowsum partials per column-slice wave
    __shared__ float oRed[2][16][64];       // O reduction across the 4 column-slice waves

    const int tid  = threadIdx.x;
    const int lane = tid & 31;
    const int w    = tid >> 5;
    const int lm   = lane & 15;             // column (B/C/D) or row (A) within tile
    const int h    = lane >> 4;             // half-wave selector
    const int mt   = w & 1;                 // row tile 0..1
    const int ct   = w >> 1;                // column slice 0..3
    const int qbase = blockIdx.x * 32;
    const int rowQ  = qbase + 16 * mt + lm;

    // zero the O reduction buffer
    for (int i = tid; i < 2 * 16 * 64; i += 256) ((float*)oRed)[i] = 0.f;

    // Preload this wave's Q fragments (A-operand layout, 16 k-steps of K=4).
    v2f qa[16];
#pragma unroll
    for (int i = 0; i < 16; ++i)
        qa[i] = *(const v2f*)(Q + rowQ * DIM + 4 * i + 2 * h);

    v8f o[4];
#pragma unroll
    for (int n = 0; n < 4; ++n) o[n] = (v8f){0.f,0.f,0.f,0.f,0.f,0.f,0.f,0.f};
    float rs[8] = {0.f,0.f,0.f,0.f,0.f,0.f,0.f,0.f};

    __syncthreads();   // oRed zeroed before phase-A atomics

    // ---------------- Phase A: rowsums + O accumulation ----------------
    for (int t = 0; t < SEQ / 64; ++t) {
        const int cb = t * 64 + ct * 16;
        const float* Krow = K + (cb + lm) * DIM + 2 * h;

        v8f s = (v8f){0.f,0.f,0.f,0.f,0.f,0.f,0.f,0.f};
#pragma unroll
        for (int i = 0; i < 16; ++i) {
            v2f b = *(const v2f*)(Krow + 4 * i);
            s = __builtin_amdgcn_wmma_f32_16x16x4_f32(false, qa[i], false, b,
                                                      (short)0, s, false, false);
        }

        float e[8];
#pragma unroll
        for (int g = 0; g < 8; ++g) { e[g] = __expf(s[g]); rs[g] += e[g]; }

        // C-layout -> A-layout transpose through wave-private LDS
#pragma unroll
        for (int g = 0; g < 8; ++g) eT[w][(g + 8 * h) * 18 + lm] = e[g];
        asm volatile("s_wait_dscnt 0" ::: "memory");
        v2f ea[4];
#pragma unroll
        for (int i = 0; i < 4; ++i)
            ea[i] = *(const v2f*)&eT[w][lm * 18 + 4 * i + 2 * h];

        // O += E @ V   (K-dim = 16 tile columns -> 4 WMMA steps, 4 output chunks)
#pragma unroll
        for (int n = 0; n < 4; ++n) {
#pragma unroll
            for (int i = 0; i < 4; ++i) {
                const int kr = cb + 4 * i + 2 * h;
                v2f b;
                b.x = V[kr * DIM + n * 16 + lm];
                b.y = V[(kr + 1) * DIM + n * 16 + lm];
                o[n] = __builtin_amdgcn_wmma_f32_16x16x4_f32(false, ea[i], false, b,
                                                             (short)0, o[n], false, false);
            }
        }
    }

    // ---------------- Reductions ----------------
#pragma unroll
    for (int g = 0; g < 8; ++g) {
        rs[g] += __shfl_xor(rs[g], 1, 32);
        rs[g] += __shfl_xor(rs[g], 2, 32);
        rs[g] += __shfl_xor(rs[g], 4, 32);
        rs[g] += __shfl_xor(rs[g], 8, 32);
    }
    if (lm == 0) {
#pragma unroll
        for (int g = 0; g < 8; ++g) rsWG[ct][mt * 16 + g + 8 * h] = rs[g];
    }
#pragma unroll
    for (int n = 0; n < 4; ++n)
#pragma unroll
        for (int g = 0; g < 8; ++g)
            atomicAdd(&oRed[mt][g + 8 * h][n * 16 + lm], o[n][g]);
    __syncthreads();

    float inv[8];
#pragma unroll
    for (int g = 0; g < 8; ++g) {
        const int r = mt * 16 + g + 8 * h;
        inv[g] = 1.0f / (rsWG[0][r] + rsWG[1][r] + rsWG[2][r] + rsWG[3][r]);
    }

    if (ct == 0) {
#pragma unroll
        for (int n = 0; n < 4; ++n)
#pragma unroll
            for (int g = 0; g < 8; ++g) {
                const int r = qbase + mt * 16 + g + 8 * h;
                Out[r * DIM + n * 16 + lm] = oRed[mt][g + 8 * h][n * 16 + lm] * inv[g];
            }
    }

    // ---------------- Phase B: write normalized scores (single streaming pass) ----------------
    for (int t = 0; t < SEQ / 64; ++t) {
        const int cb = t * 64 + ct * 16;
        const float* Krow = K + (cb + lm) * DIM + 2 * h;

        v8f s = (v8f){0.f,0.f,0.f,0.f,0.f,0.f,0.f,0.f};
#pragma unroll
        for (int i = 0; i < 16; ++i) {
            v2f b = *(const v2f*)(Krow + 4 * i);
            s = __builtin_amdgcn_wmma_f32_16x16x4_f32(false, qa[i], false, b,
                                                      (short)0, s, false, false);
        }
#pragma unroll
        for (int g = 0; g < 8; ++g) {
            const size_t r = (size_t)(qbase + mt * 16 + g + 8 * h);
            const float p = __expf(s[g]) * inv[g];
            __builtin_nontemporal_store(p, Score + r * SEQ + cb + lm);
        }
    }
}

extern "C" void kernel_launch(void* const* d_in, const int* in_sizes, int n_in,
                              void* d_out, int out_size, void* d_ws, size_t ws_size,
                              hipStream_t stream) {
    (void)in_sizes; (void)n_in; (void)d_ws; (void)ws_size; (void)out_size;
    const float* Q = (const float*)d_in[0];
    const float* K = (const float*)d_in[1];
    const float* V = (const float*)d_in[2];
    float* Out   = (float*)d_out;                       // [8192, 64]
    float* Score = (float*)d_out + (size_t)SEQ * DIM;   // [8192, 8192]
    attn_fused_kernel<<<SEQ / 32, 256, 0, stream>>>(Q, K, V, Out, Score);
}